// Net_21174188769584
// MI455X (gfx1250) — compile-verified
//
#include <hip/hip_runtime.h>
#include <hip/hip_bf16.h>

typedef __attribute__((ext_vector_type(2))) float v2f;
typedef __attribute__((ext_vector_type(8))) float v8f;

// Region boundaries (inclusive), covering nodes 0..67 contiguously.
__constant__ int c_rs[9] = {0, 17, 22, 27, 31, 36, 42, 48, 60};
__constant__ int c_re[9] = {16, 21, 26, 30, 35, 41, 47, 59, 67};

#define NODES_IN   68
#define NODES_OUT  77
#define D          128
#define ROW77      (NODES_OUT * D)   // 9856 floats per (b,t)
#define ROW68      (NODES_IN * D)    // 8704 floats per (b,t)
#define ADJS       84                // padded adj LDS row stride (bank-conflict-free A reads)
#define AGS        132               // padded agg LDS row stride (bank-conflict-free A reads)

// ---------------------------------------------------------------------------
// Fully fused network: one block (256 thr = 8 wave32) per (b,t) does
//   region pooling -> 4 x (adj@x@W + b, relu, residual) -> LayerNorm
// entirely in LDS; global traffic = read lm once + write out once.
// f32 WMMA 16x16x4. 40 tiles (5 mt x 8 nt); wave w owns column strip nt=w,
// mt=0..4 -> B fragment loaded once per K-step, reused by 5 WMMAs.
// ---------------------------------------------------------------------------
__global__ __launch_bounds__(256) void gcn_fused_kernel(
    const float* __restrict__ lm, float* __restrict__ out,
    const float* __restrict__ adj,
    const float* __restrict__ W0, const float* __restrict__ b0,
    const float* __restrict__ W1, const float* __restrict__ b1,
    const float* __restrict__ W2, const float* __restrict__ b2,
    const float* __restrict__ W3, const float* __restrict__ b3,
    const float* __restrict__ Wr, const float* __restrict__ br,
    const float* __restrict__ gamma, const float* __restrict__ beta)
{
    __shared__ __align__(16) float xs[80 * D];     // x (B op GEMM1, residual), layer-resident
    __shared__ __align__(16) float ag[80 * AGS];   // agg (A op GEMM2); final-layer out staging
    __shared__ float adjs[80 * ADJS];              // zero-padded adjacency (A op GEMM1)
    __shared__ float sw[NODES_IN];                 // pooling scores -> weights
    __shared__ float lnmu[80], lnrs[80];

    const int bt  = blockIdx.x;
    const int tid = threadIdx.x;

    const int w   = tid >> 5;   // wave 0..7  (= nt column strip)
    const int l   = tid & 31;
    const int m16 = l & 15;     // A row / B,D column within tile
    const int kh  = l >> 4;     // which K-pair within group of 4
    const int col = w * 16 + m16;

    // ---- stage lm rows 0..67 into LDS (b128), zero rows 68..79 ----
    {
        float4* xs4 = (float4*)xs;
        const float4 z4 = make_float4(0.f, 0.f, 0.f, 0.f);
        const float4* in4 = (const float4*)(lm + (size_t)bt * ROW68);
        for (int i = tid; i < 68 * 32; i += 256) xs4[i] = in4[i];
        for (int i = 68 * 32 + tid; i < 80 * 32; i += 256) xs4[i] = z4;
    }
    // ---- stage adj zero-padded into [80][ADJS] (once for all 4 layers) ----
    for (int i = tid; i < 80 * 80; i += 256) {
        int r = i / 80, c = i - r * 80;
        adjs[r * ADJS + c] = (r < NODES_OUT && c < NODES_OUT) ? adj[r * NODES_OUT + c] : 0.f;
    }
    __syncthreads();

    // ---- region attention pooling -> xs rows 68..76 ----
    {
        const float wr0 = Wr[l], wr1 = Wr[l + 32], wr2 = Wr[l + 64], wr3 = Wr[l + 96];
        const float br0 = br[0];
        for (int n = w; n < NODES_IN; n += 8) {
            const float* row = &xs[n * D];
            float p = row[l] * wr0 + row[l + 32] * wr1 + row[l + 64] * wr2 + row[l + 96] * wr3;
            #pragma unroll
            for (int off = 16; off > 0; off >>= 1) p += __shfl_xor(p, off, 32);
            if (l == 0) sw[n] = p + br0;
        }
        __syncthreads();
        if (tid < 9) {  // per-region softmax (disjoint ranges)
            int s0 = c_rs[tid], e0 = c_re[tid];
            float mx = -3.0e38f;
            for (int n = s0; n <= e0; ++n) mx = fmaxf(mx, sw[n]);
            float den = 0.f;
            for (int n = s0; n <= e0; ++n) den += expf(sw[n] - mx);
            float inv = 1.f / den;
            for (int n = s0; n <= e0; ++n) sw[n] = expf(sw[n] - mx) * inv;
        }
        __syncthreads();
        for (int idx = tid; idx < 9 * D; idx += 256) {  // weighted pooling
            int r = idx >> 7, d = idx & 127;
            int s0 = c_rs[r], e0 = c_re[r];
            float acc = 0.f;
            for (int n = s0; n <= e0; ++n) acc += sw[n] * xs[n * D + d];
            xs[(NODES_IN + r) * D + d] = acc;
        }
        __syncthreads();
    }

    // ---- 4 GCN layers, x resident in LDS ----
    #pragma unroll 1
    for (int layer = 0; layer < 4; ++layer) {
        const float* Wl;
        const float* bl;
        if (layer == 0)      { Wl = W0; bl = b0; }
        else if (layer == 1) { Wl = W1; bl = b1; }
        else if (layer == 2) { Wl = W2; bl = b2; }
        else                 { Wl = W3; bl = b3; }

        // GEMM1: ag[80][128] = adjs[80][80] @ xs[80][128]
        {
            v8f c[5];
            #pragma unroll
            for (int mt = 0; mt < 5; ++mt) c[mt] = (v8f){0.f,0.f,0.f,0.f,0.f,0.f,0.f,0.f};
            const float* arow[5];
            #pragma unroll
            for (int mt = 0; mt < 5; ++mt) arow[mt] = &adjs[(mt * 16 + m16) * ADJS + kh * 2];
            const float* bp = &xs[kh * 2 * D + col];
            #pragma unroll
            for (int kk = 0; kk < 20; ++kk) {
                v2f b = {bp[kk * 4 * D], bp[kk * 4 * D + D]};
                #pragma unroll
                for (int mt = 0; mt < 5; ++mt) {
                    v2f a = {arow[mt][kk * 4], arow[mt][kk * 4 + 1]};
                    c[mt] = __builtin_amdgcn_wmma_f32_16x16x4_f32(false, a, false, b,
                                                                  (short)0, c[mt], false, false);
                }
            }
            #pragma unroll
            for (int mt = 0; mt < 5; ++mt) {
                float* agw = &ag[(mt * 16 + kh * 8) * AGS + col];
                #pragma unroll
                for (int v = 0; v < 8; ++v) agw[v * AGS] = c[mt][v];
            }
        }
        __syncthreads();   // ag ready for all waves

        // GEMM2: out[80][128] = ag[80][128] @ Wl[128][128]
        v8f acc[5];
        {
            #pragma unroll
            for (int mt = 0; mt < 5; ++mt) acc[mt] = (v8f){0.f,0.f,0.f,0.f,0.f,0.f,0.f,0.f};
            const float* arow[5];
            #pragma unroll
            for (int mt = 0; mt < 5; ++mt) arow[mt] = &ag[(mt * 16 + m16) * AGS + kh * 2];
            const float* wp = &Wl[kh * 2 * D + col];
            #pragma unroll
            for (int kk = 0; kk < 32; ++kk) {
                v2f b = {wp[kk * 4 * D], wp[kk * 4 * D + D]};
                #pragma unroll
                for (int mt = 0; mt < 5; ++mt) {
                    v2f a = {arow[mt][kk * 4], arow[mt][kk * 4 + 1]};
                    acc[mt] = __builtin_amdgcn_wmma_f32_16x16x4_f32(false, a, false, b,
                                                                    (short)0, acc[mt], false, false);
                }
            }
        }
        __syncthreads();   // all waves done reading ag

        // epilogue: bias + relu (+ residual).
        // layers 0..2 -> write back into xs (same-thread RMW on unique elements);
        // layer 3     -> stage into ag for LayerNorm.
        {
            const float bcol = bl[col];
            #pragma unroll
            for (int mt = 0; mt < 5; ++mt) {
                float* xsr = &xs[(mt * 16 + kh * 8) * D + col];
                float* agw = &ag[(mt * 16 + kh * 8) * AGS + col];
                #pragma unroll
                for (int v = 0; v < 8; ++v) {
                    float val = acc[mt][v] + bcol;
                    val = fmaxf(val, 0.f);
                    if (layer < 3) xsr[v * D] = val + xsr[v * D];
                    else           agw[v * AGS] = val;
                }
            }
        }
        __syncthreads();   // new x (or LN staging) visible to all waves
    }

    // ---- LayerNorm from ag -> out ----
    if (tid < NODES_OUT) {
        const float* rowp = &ag[tid * AGS];
        float s = 0.f, s2 = 0.f;
        for (int d = 0; d < D; ++d) { float v = rowp[d]; s += v; s2 += v * v; }
        float mu  = s * (1.f / 128.f);
        float var = s2 * (1.f / 128.f) - mu * mu;
        lnmu[tid] = mu;
        lnrs[tid] = rsqrtf(var + 1e-5f);
    }
    __syncthreads();
    {
        float4* out4 = (float4*)(out + (size_t)bt * ROW77);
        for (int i = tid; i < NODES_OUT * 32; i += 256) {   // 2464 float4s
            int r = i >> 5, c4 = (i & 31) << 2;
            const float* src = &ag[r * AGS + c4];
            float mu = lnmu[r], rs = lnrs[r];
            float4 g  = ((const float4*)gamma)[i & 31];
            float4 bb = ((const float4*)beta)[i & 31];
            float4 o;
            o.x = (src[0] - mu) * rs * g.x + bb.x;
            o.y = (src[1] - mu) * rs * g.y + bb.y;
            o.z = (src[2] - mu) * rs * g.z + bb.z;
            o.w = (src[3] - mu) * rs * g.w + bb.w;
            out4[i] = o;
        }
    }
}

// ---------------------------------------------------------------------------
extern "C" void kernel_launch(void* const* d_in, const int* in_sizes, int n_in,
                              void* d_out, int out_size, void* d_ws, size_t ws_size,
                              hipStream_t stream) {
    const float* lm    = (const float*)d_in[0];
    const float* adj   = (const float*)d_in[1];
    const float* Wr    = (const float*)d_in[2];
    const float* br    = (const float*)d_in[3];
    const float* W0    = (const float*)d_in[4];
    const float* b0    = (const float*)d_in[5];
    const float* W1    = (const float*)d_in[6];
    const float* b1    = (const float*)d_in[7];
    const float* W2    = (const float*)d_in[8];
    const float* b2    = (const float*)d_in[9];
    const float* W3    = (const float*)d_in[10];
    const float* b3    = (const float*)d_in[11];
    const float* gamma = (const float*)d_in[12];
    const float* beta  = (const float*)d_in[13];

    const int BT = 32 * 128;  // 4096 independent (b,t) positions
    gcn_fused_kernel<<<BT, 256, 0, stream>>>(lm, (float*)d_out, adj,
                                             W0, b0, W1, b1, W2, b2, W3, b3,
                                             Wr, br, gamma, beta);
}